// Column_23398981829106
// MI455X (gfx1250) — compile-verified
//
#include <hip/hip_runtime.h>
#include <hip/hip_bf16.h>

// ---------------- problem constants ----------------
constexpr int    WW   = 2048;          // WIDTH
constexpr int    DEP  = 8;             // DEPTH
constexpr int    DP1  = 9;             // DEPTH+1
constexpr int    MOD  = 242;           // 30*DEPTH+2
constexpr size_t N_STATE = (size_t)WW * DP1;          // 18432
constexpr size_t N_SA    = N_STATE * MOD;             // 4,460,544
constexpr size_t N_SG    = (size_t)WW * DEP;          // 16384
constexpr size_t N_GRAD  = (size_t)WW * WW * DEP;     // 33,554,432

// output layout (flat, return order)
constexpr size_t OFF_OUTPUT = 0;
constexpr size_t OFF_STATE  = 1;
constexpr size_t OFF_SA     = OFF_STATE + N_STATE;    // 18433
constexpr size_t OFF_SG     = OFF_SA + N_SA;          // 4,478,977
constexpr size_t OFF_OG     = OFF_SG + N_SG;          // 4,495,361
constexpr size_t OFF_GRAD   = OFF_OG + N_SA;          // 8,955,905
constexpr size_t OFF_OWG    = OFF_GRAD + N_GRAD;      // 42,510,337

// workspace layout (floats)
constexpr size_t WS_HID     = 0;                      // 2048*8
constexpr size_t WS_STATE   = WS_HID + N_SG;          // 2048*9
constexpr size_t WS_CONTRIB = WS_STATE + N_STATE;     // 2048*7
constexpr size_t WS_SGR     = WS_CONTRIB + (size_t)WW * 7;
constexpr size_t WS_APV     = WS_SGR + N_SG;          // 2048*8

typedef __attribute__((ext_vector_type(2))) float v2f;
typedef __attribute__((ext_vector_type(8))) float v8f;

__device__ __forceinline__ int pymod(int a, int m) { int r = a % m; return r < 0 ? r + m : r; }

// ---------------- K1: hidden[w,d] = sum_i W[i,w,d]*state[i,d] ----------------
// 64 blocks x 256 thr; thread owns (w,d); coalesced: consecutive tid -> consecutive (w,d).
__global__ void k_hidden(const float* __restrict__ Wt, const float* __restrict__ st,
                         float* __restrict__ hid) {
    const int tid = threadIdx.x;
    const int d = tid & 7;
    const int w = blockIdx.x * 32 + (tid >> 3);
    const float* wp = Wt + (size_t)w * 8 + d;
    float acc = 0.f;
    for (int i = 0; i < WW; ++i) {
        if (i + 16 < WW) __builtin_prefetch(wp + (size_t)(i + 16) * (WW * 8), 0, 0);
        acc = fmaf(wp[(size_t)i * (WW * 8)], st[i * DP1 + d], acc);
    }
    hid[w * DEP + d] = acc;
}

// ---------------- K2: contrib[i,a] = sum_j W[i,j,a+1]*g[j,a+1]  (a=0..6) ----------------
// one block per i row; weights row re-read should hit the 192MB L2 after K1.
__global__ void k_contrib(const float* __restrict__ Wt, const float* __restrict__ g,
                          float* __restrict__ contrib) {
    __shared__ float red[256];
    const int tid = threadIdx.x;
    const int d  = tid & 7;        // depth slot
    const int jl = tid >> 3;       // 0..31
    const int i  = blockIdx.x;
    const float* wrow = Wt + (size_t)i * (WW * 8);
    float part = 0.f;
    for (int it = 0; it < WW / 32; ++it) {
        const int j = it * 32 + jl;
        part = fmaf(wrow[j * 8 + d], g[j * 8 + d], part);
    }
    red[tid] = part;
    __syncthreads();
    for (int s = 16; s > 0; s >>= 1) {
        if (jl < s) red[tid] += red[tid + s * 8];
        __syncthreads();
    }
    if (jl == 0 && d >= 1) contrib[(size_t)i * 7 + (d - 1)] = red[tid];
}

// ---------------- K3a: state' = [x | relu(hidden)]; also owg output ----------------
__global__ void k_state(const float* __restrict__ x, const float* __restrict__ hid,
                        float* __restrict__ ws_state, float* __restrict__ out) {
    const int idx = blockIdx.x * 256 + threadIdx.x;    // < 18432
    const int w = idx / DP1, c = idx - w * DP1;
    const float v = (c == 0) ? x[w] : fmaxf(hid[w * DEP + (c - 1)], 0.f);
    ws_state[idx]        = v;
    out[OFF_STATE + idx] = v;
    out[OFF_OWG + idx]   = v;
}

// ---------------- K3b: output = sum(ow * state') (single block, deterministic) ----------------
__global__ void k_output(const float* __restrict__ ws_state, const float* __restrict__ ow,
                         float* __restrict__ out) {
    __shared__ float red[256];
    float s = 0.f;
    for (int idx = threadIdx.x; idx < (int)N_STATE; idx += 256)
        s = fmaf(ws_state[idx], ow[idx], s);
    red[threadIdx.x] = s;
    __syncthreads();
    for (int st = 128; st > 0; st >>= 1) {
        if (threadIdx.x < st) red[threadIdx.x] += red[threadIdx.x + st];
        __syncthreads();
    }
    if (threadIdx.x == 0) out[OFF_OUTPUT] = red[0];
}

// ---------------- K4: sg, sgr = sg*relu_g, act_prev ----------------
__global__ void k_sg(const float* __restrict__ ws_state, const float* __restrict__ contrib,
                     const float* __restrict__ ow, const float* __restrict__ sa_in,
                     const float* __restrict__ og_in, const int* __restrict__ timep,
                     float* __restrict__ out, float* __restrict__ ws_sgr,
                     float* __restrict__ ws_apv) {
    const int idx = blockIdx.x * 256 + threadIdx.x;    // < 16384
    const int i = idx >> 3, aa = idx & 7;
    const int T = timep[0];
    const int tia = pymod(T - 2 * DEP + 2 * (aa + 1), MOD);

    auto sa_eff = [&](int w, int c, int t) -> float {
        return (t == T) ? ws_state[w * DP1 + c] : sa_in[((size_t)w * DP1 + c) * MOD + t];
    };
    auto og_eff = [&](int w, int c, int t) -> float {
        return (t == T) ? ow[w * DP1 + c] : og_in[((size_t)w * DP1 + c) * MOD + t];
    };

    float sgv;
    if (aa < 7) {
        const float rm = (sa_eff(i, aa + 2, (tia + 1) % MOD) > 0.f) ? 1.f : 0.f;
        sgv = fmaf(rm, contrib[(size_t)i * 7 + aa], og_eff(i, aa + 1, tia));
    } else {
        sgv = og_eff(i, DEP, T);   // == ow[i, DEP]
    }
    const float rg = (sa_eff(i, aa + 1, tia) > 0.f) ? 1.f : 0.f;
    const float ap = sa_eff(i, aa, pymod(tia - 1, MOD));

    out[OFF_SG + idx] = sgv;
    ws_sgr[idx] = sgv * rg;
    ws_apv[idx] = ap;
}

// ---------------- K5: sa / og outputs (stored arrays with time-slice replaced) ----------------
__global__ void k_copy(const float* __restrict__ ws_state, const float* __restrict__ ow,
                       const float* __restrict__ sa_in, const float* __restrict__ og_in,
                       const int* __restrict__ timep, float* __restrict__ out) {
    const size_t idx = (size_t)blockIdx.x * 256 + threadIdx.x;   // < N_SA
    const int wc = (int)(idx / MOD);
    const int t  = (int)(idx - (size_t)wc * MOD);
    const int T  = timep[0];
    out[OFF_SA + idx] = (t == T) ? ws_state[wc] : sa_in[idx];
    out[OFF_OG + idx] = (t == T) ? ow[wc]       : og_in[idx];
}

// ---------------- K6: gradients[i,j,aa] = act_prev[i,aa]*sgr[j,aa] via WMMA ----------------
// One wave per 16x16 (i,j) tile; 8 x V_WMMA_F32_16X16X4_F32 (K=0 column/row only -> outer product).
__global__ void k_grad(const float* __restrict__ apv, const float* __restrict__ sgr,
                       float* __restrict__ outg) {
    const int wave = (blockIdx.x * 256 + threadIdx.x) >> 5;  // 0..16383
    const int lane = threadIdx.x & 31;
    const int ti = wave >> 7;        // i tile 0..127
    const int tj = wave & 127;       // j tile 0..127
    const int r16 = lane & 15;
    const bool lo = lane < 16;

    // A needs: lanes 0-15 VGPR0 = act_prev[M=lane], rest 0 (K=1..3 zero, lanes 16-31 zero).
    // B needs: lanes 0-15 VGPR0 = sgr[N=lane] (K=0 row), rest 0.
    float a[8], b[8];
    const float* ap = apv + (size_t)(ti * 16 + r16) * DEP;
    const float* bp = sgr + (size_t)(tj * 16 + r16) * DEP;
#pragma unroll
    for (int k = 0; k < 8; ++k) {
        a[k] = lo ? ap[k] : 0.f;
        b[k] = lo ? bp[k] : 0.f;
    }

    const int iBase = ti * 16 + (lane >> 4) * 8;   // D layout: VGPR r -> M = r (+8 for hi lanes)
    const int j     = tj * 16 + r16;               // N = lane & 15

#if defined(__gfx1250__) && __has_builtin(__builtin_amdgcn_wmma_f32_16x16x4_f32)
    v8f acc[8];
#pragma unroll
    for (int k = 0; k < 8; ++k) {
        v2f av = {a[k], 0.f};
        v2f bv = {b[k], 0.f};
        v8f c  = {};
        acc[k] = __builtin_amdgcn_wmma_f32_16x16x4_f32(
            /*neg_a=*/false, av, /*neg_b=*/false, bv,
            /*c_mod=*/(short)0, c, /*reuse_a=*/false, /*reuse_b=*/false);
    }
    // Regroup per (i,j): lane holds all 8 aa values for each of its 8 M rows -> contiguous 32B runs.
#pragma unroll
    for (int r = 0; r < 8; ++r) {
        float* base = outg + ((size_t)(iBase + r) * WW + j) * DEP;
#pragma unroll
        for (int k = 0; k < 8; ++k)
            __builtin_nontemporal_store(acc[k][r], base + k);
    }
#else
    // VALU fallback, same tiling/mapping.
#pragma unroll
    for (int r = 0; r < 8; ++r) {
        float* base = outg + ((size_t)(iBase + r) * WW + j) * DEP;
        const float* arow = apv + (size_t)(iBase + r) * DEP;
#pragma unroll
        for (int k = 0; k < 8; ++k)
            __builtin_nontemporal_store(arow[k] * sgr[(size_t)j * DEP + k], base + k);
    }
#endif
}

// ---------------- host launcher ----------------
extern "C" void kernel_launch(void* const* d_in, const int* in_sizes, int n_in,
                              void* d_out, int out_size, void* d_ws, size_t ws_size,
                              hipStream_t stream) {
    const float* x     = (const float*)d_in[0];
    const float* Wt    = (const float*)d_in[1];   // (2048, 2048, 8)
    const float* ow    = (const float*)d_in[2];   // (2048, 9)
    const float* st    = (const float*)d_in[3];   // (2048, 9)
    const float* sa_in = (const float*)d_in[4];   // (2048, 9, 242)
    const float* g_in  = (const float*)d_in[5];   // (2048, 8)
    const float* og_in = (const float*)d_in[6];   // (2048, 9, 242)
    const int*   timep = (const int*)  d_in[7];

    float* out = (float*)d_out;
    float* wsf = (float*)d_ws;

    float* ws_hid     = wsf + WS_HID;
    float* ws_state   = wsf + WS_STATE;
    float* ws_contrib = wsf + WS_CONTRIB;
    float* ws_sgr     = wsf + WS_SGR;
    float* ws_apv     = wsf + WS_APV;

    k_hidden <<<WW / 32, 256, 0, stream>>>(Wt, st, ws_hid);
    k_contrib<<<WW,      256, 0, stream>>>(Wt, g_in, ws_contrib);        // weights from L2
    k_state  <<<(int)(N_STATE / 256), 256, 0, stream>>>(x, ws_hid, ws_state, out);
    k_output <<<1,       256, 0, stream>>>(ws_state, ow, out);
    k_sg     <<<(int)(N_SG / 256), 256, 0, stream>>>(ws_state, ws_contrib, ow, sa_in,
                                                     og_in, timep, out, ws_sgr, ws_apv);
    k_copy   <<<(int)(N_SA / 256), 256, 0, stream>>>(ws_state, ow, sa_in, og_in, timep, out);
    k_grad   <<<(WW / 16) * (WW / 16) / 8, 256, 0, stream>>>(ws_apv, ws_sgr, out + OFF_GRAD);
}